// UnetGraphSAGE_29326036697880
// MI455X (gfx1250) — compile-verified
//
#include <hip/hip_runtime.h>

// ---------------------------------------------------------------------------
// CDNA5 (gfx1250) R-GCN U-Net on the cube-sphere.
// GEMMs run on the matrix pipe via V_WMMA_F32_16X16X4_F32 (the only fp32 WMMA
// shape -> bit-faithful fp32 math). One wave owns a 16x16 node x channel tile
// and accumulates all 3 relations at once (A-tile reuse, 3 independent WMMA
// chains). FOUT is a compile-time template parameter so all B-tile addressing
// folds into immediate offsets + constant pointer bumps. Edge scatter =
// contiguous global_atomic_add_f32 (the HBM roofline limiter) with shift-only
// addressing (all fout are pow2).
// ---------------------------------------------------------------------------

#define R_REL 3
#define B_BASES 2
#define DEG 8

typedef float v2f __attribute__((ext_vector_type(2)));
typedef float v8f __attribute__((ext_vector_type(8)));

static inline int cdiv_i(long long a, int b) { return (int)((a + b - 1) / b); }
static inline int ilog2_i(int v) { int s = 0; while ((1 << s) < v) ++s; return s; }

// --- 1. basis combine, faithful to the reference's reshape aliasing --------
// Writes tmp[f, r, o] (flat f*R*fout + r*fout + o); the GEMM later reads the
// SAME flat buffer as w[r, i, o] (flat r*fin*fout + i*fout + o).
__global__ void relw_kernel(const float* __restrict__ W, const float* __restrict__ wc,
                            float* __restrict__ w, int fin, int fout) {
  long long idx = blockIdx.x * (long long)blockDim.x + threadIdx.x;
  long long total = (long long)fin * R_REL * fout;
  if (idx >= total) return;
  int o = (int)(idx % fout);
  long long t = idx / fout;
  int r = (int)(t % R_REL);
  int f = (int)(t / R_REL);
  float acc = 0.f;
#pragma unroll
  for (int b = 0; b < B_BASES; ++b)
    acc += wc[r * B_BASES + b] * W[((long long)f * B_BASES + b) * fout + o];
  w[idx] = acc;
}

// --- 2. proj[n, r, o] = sum_i h[n, i] * w[r, i, o]  via fp32 WMMA ----------
// One wave per (16-node tile) x (16-col tile); all 3 relations accumulated
// simultaneously from one shared A-tile load.
// A layout (16x4 f32): lane L<16 -> M=L, K=k+{0,1}; lane L>=16 -> M=L-16, K=k+{2,3}.
// B layout (4x16 f32): mirrored K split, N = lane&15.
// D layout: VGPR v -> M = v + 8*(lane>>4), N = lane&15.
// Ragged FOUT (<16, only c11): loads use a clamped column (in-range), the
// resulting garbage columns are simply never stored -> branch-free inner loop,
// EXEC all-1s throughout (wave-uniform early exit only).
template <int FOUT>
__global__ void rgcn_gemm_wmma(const float* __restrict__ h, const float* __restrict__ w,
                               float* __restrict__ proj, int N, int fin) {
  constexpr int tilesO = (FOUT + 15) / 16;
  const int lane = threadIdx.x & 31;
  const int wave = (int)((blockIdx.x * (long long)blockDim.x + threadIdx.x) >> 5);
  const int total = (N >> 4) * tilesO;
  if (wave >= total) return;                 // wave-uniform: EXEC stays all-1s
  const int ot = wave % tilesO;
  const int mt = wave / tilesO;

  const int lo = lane & 15;
  const int hi = lane >> 4;
  const int hiK = hi << 1;                   // 0 for lanes 0-15, 2 for 16-31
  const int row = (mt << 4) + lo;
  const int col = (ot << 4) + lo;
  const int colc = (col < FOUT) ? col : 0;   // clamped load column

  const float* __restrict__ hr = h + (size_t)row * fin + hiK;
  const size_t rstride = (size_t)fin * FOUT;
  const float* __restrict__ w0 = w + (size_t)hiK * FOUT + colc;
  const float* __restrict__ w1 = w0 + rstride;
  const float* __restrict__ w2 = w1 + rstride;

  v8f acc0 = {}, acc1 = {}, acc2 = {};
  for (int k = 0; k < fin; k += 4) {
    v2f a, b0, b1, b2;
    a.x = hr[0];                             // pair -> global_load_b64
    a.y = hr[1];
    hr += 4;
    b0.x = w0[0]; b0.y = w0[FOUT];           // immediate-offset b32 loads
    b1.x = w1[0]; b1.y = w1[FOUT];
    b2.x = w2[0]; b2.y = w2[FOUT];
    w0 += 4 * FOUT; w1 += 4 * FOUT; w2 += 4 * FOUT;
    acc0 = __builtin_amdgcn_wmma_f32_16x16x4_f32(false, a, false, b0, (short)0, acc0, false, false);
    acc1 = __builtin_amdgcn_wmma_f32_16x16x4_f32(false, a, false, b1, (short)0, acc1, false, false);
    acc2 = __builtin_amdgcn_wmma_f32_16x16x4_f32(false, a, false, b2, (short)0, acc2, false, false);
  }
  if (col < FOUT) {
    const int mbase = (mt << 4) + (hi << 3);
    float* __restrict__ p0 = proj + (size_t)mbase * R_REL * FOUT + col;
#pragma unroll
    for (int v = 0; v < 8; ++v) {
      p0[0]        = acc0[v];
      p0[FOUT]     = acc1[v];
      p0[2 * FOUT] = acc2[v];
      p0 += R_REL * FOUT;
    }
  }
}

// --- 3. zero / scatter / relu ---------------------------------------------
__global__ void fill0_kernel(float* __restrict__ p, long long n) {
  long long i = blockIdx.x * (long long)blockDim.x + threadIdx.x;
  if (i < n) p[i] = 0.f;
}

// fout is always a power of two -> shift/mask addressing only.
__global__ void rgcn_scatter(const float* __restrict__ proj, const int* __restrict__ src,
                             const int* __restrict__ dst, const int* __restrict__ rel,
                             const float* __restrict__ norm, float* __restrict__ agg,
                             long long E, int foutSh) {
  long long idx = blockIdx.x * (long long)blockDim.x + threadIdx.x;
  long long total = E << foutSh;
  if (idx >= total) return;
  long long e = idx >> foutSh;
  int o = (int)idx & ((1 << foutSh) - 1);
  int s = src[e];
  int d = dst[e];
  float v = proj[(((size_t)s * R_REL + rel[e]) << foutSh) + o] * norm[e];
  atomicAdd(&agg[((size_t)d << foutSh) + o], v);   // global_atomic_add_f32, contiguous
}

__global__ void relu_kernel(float* __restrict__ p, long long n) {
  long long i = blockIdx.x * (long long)blockDim.x + threadIdx.x;
  if (i < n) p[i] = fmaxf(p[i], 0.f);
}

// --- 4. mean pool 2x2 per face --------------------------------------------
__global__ void pool_kernel(const float* __restrict__ in, float* __restrict__ out,
                            int r, int C) {
  const int half = r >> 1;
  long long total = (long long)6 * half * half * C;
  long long idx = blockIdx.x * (long long)blockDim.x + threadIdx.x;
  if (idx >= total) return;
  long long n = idx / C;
  int c = (int)(idx % C);
  int f = (int)(n / (half * half));
  int rem = (int)(n % (half * half));
  int y = rem / half, x = rem % half;
  long long i00 = (long long)f * r * r + (long long)(2 * y) * r + 2 * x;
  float v = in[i00 * C + c] + in[(i00 + 1) * C + c] +
            in[(i00 + r) * C + c] + in[(i00 + r + 1) * C + c];
  out[idx] = 0.25f * v;
}

// --- 5. ConvTranspose2d k=2 s=2 (Cin == Cout == C) ------------------------
__global__ void upsample_kernel(const float* __restrict__ in, const float* __restrict__ W,
                                const float* __restrict__ bias, float* __restrict__ out,
                                int r, int C) {
  const int R2 = 2 * r;
  long long total = (long long)6 * R2 * R2 * C;
  long long idx = blockIdx.x * (long long)blockDim.x + threadIdx.x;
  if (idx >= total) return;
  long long n = idx / C;
  int d = (int)(idx % C);
  int f = (int)(n / (R2 * R2));
  int rem = (int)(n % (R2 * R2));
  int Y = rem / R2, X = rem % R2;
  int y = Y >> 1, x = X >> 1, a = Y & 1, b = X & 1;
  const float* __restrict__ hin = in + ((size_t)f * r * r + (size_t)y * r + x) * C;
  const int wofs = d * 4 + a * 2 + b;
  float acc = bias[d];
  for (int c = 0; c < C; ++c)
    acc += hin[c] * W[(size_t)c * C * 4 + wofs];
  out[idx] = acc;
}

// --- 6. channel concat -----------------------------------------------------
__global__ void concat_kernel(const float* __restrict__ a, const float* __restrict__ b,
                              float* __restrict__ out, long long N, int ca, int cb) {
  const int ct = ca + cb;
  long long total = N * ct;
  long long idx = blockIdx.x * (long long)blockDim.x + threadIdx.x;
  if (idx >= total) return;
  long long n = idx / ct;
  int c = (int)(idx % ct);
  out[idx] = (c < ca) ? a[n * ca + c] : b[n * cb + (c - ca)];
}

// ---------------------------------------------------------------------------
// Host orchestration
// ---------------------------------------------------------------------------
namespace {

struct Graph {
  const int* src; const int* dst; const int* rel; const float* norm; int N;
};

void launch_gemm(hipStream_t stream, const float* h, const float* w, float* proj,
                 int N, int fin, int fout) {
  const int TB = 256;
  const int tilesO = (fout + 15) >> 4;
  const long long waves = (long long)(N >> 4) * tilesO;
  const int blocks = cdiv_i(waves * 32, TB);
  switch (fout) {
    case 4:   rgcn_gemm_wmma<4>  <<<blocks, TB, 0, stream>>>(h, w, proj, N, fin); break;
    case 32:  rgcn_gemm_wmma<32> <<<blocks, TB, 0, stream>>>(h, w, proj, N, fin); break;
    case 64:  rgcn_gemm_wmma<64> <<<blocks, TB, 0, stream>>>(h, w, proj, N, fin); break;
    case 128: rgcn_gemm_wmma<128><<<blocks, TB, 0, stream>>>(h, w, proj, N, fin); break;
    case 256: rgcn_gemm_wmma<256><<<blocks, TB, 0, stream>>>(h, w, proj, N, fin); break;
    case 512: rgcn_gemm_wmma<512><<<blocks, TB, 0, stream>>>(h, w, proj, N, fin); break;
  }
}

void run_rgcn(hipStream_t stream, const float* hin, int N, int fin, int fout,
              const float* W, const float* wc, const Graph& g,
              float* hout, float* proj, float* wbuf) {
  const int TB = 256;
  long long nw = (long long)fin * R_REL * fout;
  relw_kernel<<<cdiv_i(nw, TB), TB, 0, stream>>>(W, wc, wbuf, fin, fout);

  launch_gemm(stream, hin, wbuf, proj, N, fin, fout);

  long long nagg = (long long)N * fout;
  fill0_kernel<<<cdiv_i(nagg, TB), TB, 0, stream>>>(hout, nagg);

  long long E = (long long)N * DEG;
  rgcn_scatter<<<cdiv_i(E * fout, TB), TB, 0, stream>>>(
      proj, g.src, g.dst, g.rel, g.norm, hout, E, ilog2_i(fout));

  relu_kernel<<<cdiv_i(nagg, TB), TB, 0, stream>>>(hout, nagg);
}

}  // namespace

extern "C" void kernel_launch(void* const* d_in, const int* in_sizes, int n_in,
                              void* d_out, int out_size, void* d_ws, size_t ws_size,
                              hipStream_t stream) {
  (void)in_sizes; (void)n_in; (void)out_size; (void)ws_size;
  const int TB = 256;
  static const int NS[5] = {98304, 24576, 6144, 1536, 384};

  const float* x = (const float*)d_in[0];
  Graph g[5];
  for (int l = 0; l < 5; ++l) {
    int base = 1 + 4 * l;
    g[l].src  = (const int*)d_in[base + 0];
    g[l].dst  = (const int*)d_in[base + 1];
    g[l].rel  = (const int*)d_in[base + 2];
    g[l].norm = (const float*)d_in[base + 3];
    g[l].N = NS[l];
  }
  // params: LAYERS order c1,c2,c3,c33,c4,c44,c5,c55,c6,c66,c7,c77,c8,c88,c9,c99,c10,c101,c11
  enum { C1, C2, C3, C33, C4, C44, C5, C55, C6, C66, C7, C77, C8, C88, C9, C99, C10, C101, C11 };
  const float* LW[19]; const float* LC[19];
  for (int i = 0; i < 19; ++i) {
    LW[i] = (const float*)d_in[21 + 2 * i];
    LC[i] = (const float*)d_in[21 + 2 * i + 1];
  }
  const float* upW[4]; const float* upB[4];
  for (int i = 0; i < 4; ++i) {
    upW[i] = (const float*)d_in[59 + 2 * i];
    upB[i] = (const float*)d_in[59 + 2 * i + 1];
  }

  // workspace carve-up (floats)
  float* ws = (float*)d_ws;
  size_t off = 0;
  float* bufA   = ws + off; off += (size_t)98304 * 64;         // ping
  float* bufB   = ws + off; off += (size_t)98304 * 64;         // pong
  float* proj   = ws + off; off += (size_t)98304 * R_REL * 32; // max N*R*fout
  float* wbuf   = ws + off; off += (size_t)R_REL * 512 * 512;
  float* skip22 = ws + off; off += (size_t)98304 * 32;
  float* skip33 = ws + off; off += (size_t)24576 * 64;
  float* skip44 = ws + off; off += (size_t)6144 * 128;
  float* skip55 = ws + off; off += (size_t)1536 * 256;

  const int N0 = NS[0], N1 = NS[1], N2 = NS[2], N3 = NS[3], N4 = NS[4];

  // ---- encoder ----
  run_rgcn(stream, x,    N0,   4,  32, LW[C1],  LC[C1],  g[0], bufA,   proj, wbuf); // h1
  run_rgcn(stream, bufA, N0,  32,  32, LW[C2],  LC[C2],  g[0], skip22, proj, wbuf); // h22
  pool_kernel<<<cdiv_i((long long)N1 * 32, TB), TB, 0, stream>>>(skip22, bufA, 128, 32);
  run_rgcn(stream, bufA, N1,  32,  64, LW[C3],  LC[C3],  g[1], bufB,   proj, wbuf);
  run_rgcn(stream, bufB, N1,  64,  64, LW[C33], LC[C33], g[1], skip33, proj, wbuf); // h33
  pool_kernel<<<cdiv_i((long long)N2 * 64, TB), TB, 0, stream>>>(skip33, bufA, 64, 64);
  run_rgcn(stream, bufA, N2,  64, 128, LW[C4],  LC[C4],  g[2], bufB,   proj, wbuf);
  run_rgcn(stream, bufB, N2, 128, 128, LW[C44], LC[C44], g[2], skip44, proj, wbuf); // h44
  pool_kernel<<<cdiv_i((long long)N3 * 128, TB), TB, 0, stream>>>(skip44, bufA, 32, 128);
  run_rgcn(stream, bufA, N3, 128, 256, LW[C5],  LC[C5],  g[3], bufB,   proj, wbuf);
  run_rgcn(stream, bufB, N3, 256, 256, LW[C55], LC[C55], g[3], skip55, proj, wbuf); // h55
  pool_kernel<<<cdiv_i((long long)N4 * 256, TB), TB, 0, stream>>>(skip55, bufA, 16, 256);

  // ---- bottleneck ----
  run_rgcn(stream, bufA, N4, 256, 512, LW[C6],  LC[C6],  g[4], bufB, proj, wbuf);
  run_rgcn(stream, bufB, N4, 512, 512, LW[C66], LC[C66], g[4], bufA, proj, wbuf);
  run_rgcn(stream, bufA, N4, 512, 256, LW[C7],  LC[C7],  g[4], bufB, proj, wbuf);

  // ---- decoder ----
  upsample_kernel<<<cdiv_i((long long)N3 * 256, TB), TB, 0, stream>>>(bufB, upW[0], upB[0], bufA, 8, 256);
  concat_kernel<<<cdiv_i((long long)N3 * 512, TB), TB, 0, stream>>>(bufA, skip55, bufB, N3, 256, 256);
  run_rgcn(stream, bufB, N3, 512, 256, LW[C7],  LC[C7],  g[3], bufA, proj, wbuf);
  run_rgcn(stream, bufA, N3, 256, 256, LW[C77], LC[C77], g[3], bufB, proj, wbuf);
  run_rgcn(stream, bufB, N3, 256, 128, LW[C8],  LC[C8],  g[3], bufA, proj, wbuf);

  upsample_kernel<<<cdiv_i((long long)N2 * 128, TB), TB, 0, stream>>>(bufA, upW[1], upB[1], bufB, 16, 128);
  concat_kernel<<<cdiv_i((long long)N2 * 256, TB), TB, 0, stream>>>(bufB, skip44, bufA, N2, 128, 128);
  run_rgcn(stream, bufA, N2, 256, 128, LW[C8],  LC[C8],  g[2], bufB, proj, wbuf);
  run_rgcn(stream, bufB, N2, 128, 128, LW[C88], LC[C88], g[2], bufA, proj, wbuf);
  run_rgcn(stream, bufA, N2, 128,  64, LW[C9],  LC[C9],  g[2], bufB, proj, wbuf);

  upsample_kernel<<<cdiv_i((long long)N1 * 64, TB), TB, 0, stream>>>(bufB, upW[2], upB[2], bufA, 32, 64);
  concat_kernel<<<cdiv_i((long long)N1 * 128, TB), TB, 0, stream>>>(bufA, skip33, bufB, N1, 64, 64);
  run_rgcn(stream, bufB, N1, 128,  64, LW[C9],  LC[C9],  g[1], bufA, proj, wbuf);
  run_rgcn(stream, bufA, N1,  64,  64, LW[C99], LC[C99], g[1], bufB, proj, wbuf);
  run_rgcn(stream, bufB, N1,  64,  32, LW[C10], LC[C10], g[1], bufA, proj, wbuf);

  upsample_kernel<<<cdiv_i((long long)N0 * 32, TB), TB, 0, stream>>>(bufA, upW[3], upB[3], bufB, 64, 32);
  concat_kernel<<<cdiv_i((long long)N0 * 64, TB), TB, 0, stream>>>(bufB, skip22, bufA, N0, 32, 32);
  run_rgcn(stream, bufA, N0,  64,  32, LW[C10],  LC[C10],  g[0], bufB, proj, wbuf);
  run_rgcn(stream, bufB, N0,  32,  32, LW[C101], LC[C101], g[0], bufA, proj, wbuf);
  run_rgcn(stream, bufA, N0,  32,   4, LW[C11],  LC[C11],  g[0], (float*)d_out, proj, wbuf);
}